// Loss_68676527063674
// MI455X (gfx1250) — compile-verified
//
#include <hip/hip_runtime.h>

// CDNA5 / gfx1250 — wave32, WMMA (16x16xK), 320KB LDS per WGP, async LDS loads.
// SSD loss: bandwidth-bound on plabel (362MB -> ~15.5us @ 23.3TB/s).

typedef float v2f __attribute__((ext_vector_type(2)));
typedef float v8f __attribute__((ext_vector_type(8)));
typedef int   v4i __attribute__((ext_vector_type(4)));

typedef __attribute__((address_space(1))) v4i* gv4i_p;   // global (AS1)
typedef __attribute__((address_space(3))) v4i* lv4i_p;   // LDS (AS3)

#define TPB 256
#define NEG_FLT_MAX (-3.402823466e+38f)

// --- gfx1250 async global->LDS path (guarded: falls back to direct loads) ---
#if defined(__has_builtin)
#  if __has_builtin(__builtin_amdgcn_global_load_async_to_lds_b128)
#    define HAVE_ASYNC_LDS 1
#  endif
#endif
#ifndef HAVE_ASYNC_LDS
#  define HAVE_ASYNC_LDS 0
#endif

#if defined(__has_builtin) && __has_builtin(__builtin_amdgcn_s_wait_asynccnt)
#  define WAIT_ASYNCCNT(n) __builtin_amdgcn_s_wait_asynccnt(n)
#else
#  define WAIT_ASYNCCNT(n) asm volatile("s_wait_asynccnt %0" :: "i"(n) : "memory")
#endif

#if HAVE_ASYNC_LDS
__device__ __forceinline__ void async_copy_b128(const float* gsrc, float* ldst) {
    // Builtin signature (from hipcc diagnostic): param0 = v4i __device__* (AS1),
    // non-const; order = (global src, lds dst, imm offset, imm cpol).
    __builtin_amdgcn_global_load_async_to_lds_b128(
        (gv4i_p)reinterpret_cast<v4i*>(const_cast<float*>(gsrc)),
        (lv4i_p)reinterpret_cast<v4i*>(ldst),
        0, 0);
}
#endif

// ---------------------------------------------------------------------------
// Deterministic block reduction (blockDim.x == 256)
__device__ __forceinline__ float blockReduceSum(float v, float* s_red) {
    int t = threadIdx.x;
    s_red[t] = v;
    __syncthreads();
    for (int o = 128; o > 0; o >>= 1) {
        if (t < o) s_red[t] += s_red[t + o];
        __syncthreads();
    }
    float r = s_red[0];
    __syncthreads();
    return r;
}

// ---------------------------------------------------------------------------
// Kernel A: per-anchor cross entropy  con[n,a] = logsumexp_c(plabel) - x_label
// Streams plabel once. Class row c+1 is staged global->LDS asynchronously
// (ASYNCcnt loads complete in order => wait<=1 makes row c visible) while
// row c feeds the online logsumexp. Each lane consumes exactly the 16B it
// staged itself, so the per-wave counter wait is the only synchronization.
__global__ void con_kernel(const float* __restrict__ plabel,
                           const int*   __restrict__ glabel,
                           float*       __restrict__ ws_con,
                           int N, int C, int A) {
    int n   = blockIdx.x;
    int tid = threadIdx.x;
    int a0  = (blockIdx.y * TPB + tid) * 4;   // A % 4 == 0 (8732)
    if (a0 >= A) return;

    const float* base = plabel + (size_t)n * C * A;
    int   lbl[4];
    float m[4], s[4], sel[4];
#pragma unroll
    for (int j = 0; j < 4; ++j) {
        m[j] = NEG_FLT_MAX; s[j] = 0.0f; sel[j] = 0.0f;
        lbl[j] = glabel[(size_t)n * A + a0 + j];
    }

#if HAVE_ASYNC_LDS
    __shared__ __align__(16) float tile[2][TPB * 4];   // 8KB double buffer
    float* slot[2] = { &tile[0][tid * 4], &tile[1][tid * 4] };
    async_copy_b128(base + a0, slot[0]);               // stage row 0
#endif

    for (int c = 0; c < C; ++c) {
        float4 v;
#if HAVE_ASYNC_LDS
        if (c + 1 < C) {
            async_copy_b128(base + (size_t)(c + 1) * A + a0, slot[(c + 1) & 1]);
            WAIT_ASYNCCNT(1);      // row c complete (in-order async loads)
        } else {
            WAIT_ASYNCCNT(0);
        }
        asm volatile("" ::: "memory");
        v = *reinterpret_cast<const float4*>(slot[c & 1]);
#else
        v = *reinterpret_cast<const float4*>(base + (size_t)c * A + a0);
        if (c + 1 < C)   // gfx1250: lowers to global_prefetch_b8
            __builtin_prefetch((const void*)(base + (size_t)(c + 1) * A + a0), 0, 3);
#endif
        float x[4] = {v.x, v.y, v.z, v.w};
#pragma unroll
        for (int j = 0; j < 4; ++j) {
            float xv = x[j];
            if (xv <= m[j]) {                    // common case: one exp
                s[j] += __expf(xv - m[j]);
            } else {                             // new running max: rescale
                s[j] = s[j] * __expf(m[j] - xv) + 1.0f;
                m[j] = xv;
            }
            if (c == lbl[j]) sel[j] = xv;
        }
    }
#pragma unroll
    for (int j = 0; j < 4; ++j) {
        float con = m[j] + __logf(s[j]) - sel[j];   // >= 0 mathematically
        ws_con[(size_t)n * A + a0 + j] = fmaxf(con, 0.0f);
    }
}

// ---------------------------------------------------------------------------
// Kernel B: one workgroup per image.
//  - SmoothL1 loc loss: K=4 coordinate sum done with V_WMMA_F32_16X16X4_F32,
//    accumulating D across anchor blocks (D = A x ones + C).
//  - Exact hard-negative mining: radix-select threshold over con_neg float
//    bits + ordered tie scan (stable argsort semantics).
__global__ void image_kernel(const float* __restrict__ ploc,
                             const float* __restrict__ gloc,
                             const int*   __restrict__ glabel,
                             const float* __restrict__ dboxes,
                             const float* __restrict__ ws_con,
                             float* __restrict__ ws_locc,
                             float* __restrict__ ws_conc,
                             int N, int A) {
    extern __shared__ unsigned char smem[];
    float*    s_con  = (float*)smem;            // A floats
    unsigned* s_cneg = (unsigned*)(s_con + A);  // A uints (con_neg bit patterns)

    __shared__ int      s_hist[256];
    __shared__ int      s_scan[256];
    __shared__ float    s_red[256];
    __shared__ float    s_wred[16];
    __shared__ int      s_pos;
    __shared__ unsigned sh_prefix;
    __shared__ int      sh_kk;
    __shared__ int      sh_G;
    __shared__ int      sh_run;
    __shared__ float    sh_loc;

    int t = threadIdx.x;
    int n = blockIdx.x;
    if (t == 0) s_pos = 0;
    __syncthreads();

    // ---- phase 1: stage con row + masks into LDS; count positives ---------
    float sp = 0.0f;
    int   pc = 0;
    for (int i = t; i < A; i += TPB) {
        float cv  = fmaxf(ws_con[(size_t)n * A + i], 0.0f);
        bool  pos = glabel[(size_t)n * A + i] > 0;
        s_con[i]  = cv;
        s_cneg[i] = pos ? 0u : __float_as_uint(cv);   // con_neg = where(mask,0,con)
        if (pos) { pc++; sp += cv; }
    }
    atomicAdd(&s_pos, pc);                    // integer: deterministic
    float sum_pos = blockReduceSum(sp, s_red);

    // ---- phase 2: loc loss via WMMA f32 16x16x4 ----------------------------
    // A-matrix layout (16x4 f32): lanes 0-15 hold K=0,1; lanes 16-31 K=2,3.
    int w    = t >> 5;
    int lane = t & 31;
    int half = lane >> 4;
    int mrow = lane & 15;

    v8f cacc = {};
    v2f bones; bones.x = 1.0f; bones.y = 1.0f;   // ones B => D[M][*] = rowsum(A[M])

    int NB = (A + 15) >> 4;
    for (int b = w; b < NB; b += 8) {            // wave-uniform: EXEC stays full
        int  a     = b * 16 + mrow;
        bool valid = a < A;
        int  ac    = valid ? a : 0;
        int  c0    = half * 2;
        float sl[2];
#pragma unroll
        for (int j = 0; j < 2; ++j) {
            int   c = c0 + j;
            float p = ploc[((size_t)n * 4 + c) * A + ac];
            float g = gloc[((size_t)n * 4 + c) * A + ac];
            float vgd;
            if (half == 0) {   // xy coords: 10*(g - db)/db_wh
                float db  = dboxes[(size_t)c * A + ac];
                float dbw = dboxes[(size_t)(c + 2) * A + ac];
                vgd = 10.0f * (g - db) / dbw;
            } else {           // wh coords: 5*log(g/db_wh)
                float dbw = dboxes[(size_t)c * A + ac];
                vgd = 5.0f * __logf(g / dbw);
            }
            float d  = p - vgd;
            float ad = fabsf(d);
            sl[j] = (ad < 1.0f) ? (0.5f * d * d) : (ad - 0.5f);
        }
        bool  pos = valid && (glabel[(size_t)n * A + ac] > 0);
        float mf  = pos ? 1.0f : 0.0f;
        v2f amat; amat.x = sl[0] * mf; amat.y = sl[1] * mf;
        cacc = __builtin_amdgcn_wmma_f32_16x16x4_f32(
            false, amat, false, bones, (short)0, cacc, false, false);
    }
    // D replicated across columns: total = column0 of M=0..7 (lane 0) + M=8..15 (lane 16)
    float partial = cacc[0] + cacc[1] + cacc[2] + cacc[3] +
                    cacc[4] + cacc[5] + cacc[6] + cacc[7];
    if (mrow == 0) s_wred[w * 2 + half] = partial;
    __syncthreads();
    if (t == 0) {
        float ls = 0.0f;
        for (int i = 0; i < 16; ++i) ls += s_wred[i];
        sh_loc = ls;
    }
    __syncthreads();

    int posn = s_pos;
    if (posn == 0) {                 // uniform branch: whole block exits
        if (t == 0) { ws_locc[n] = 0.0f; ws_conc[n] = 0.0f; }
        return;
    }
    int k = 3 * posn; if (k > A) k = A;

    // ---- phase 3: radix-select k-th largest of con_neg (float bits) --------
    if (t == 0) { sh_prefix = 0u; sh_kk = k; sh_G = 0; }
    __syncthreads();
    for (int p = 0; p < 4; ++p) {
        int shift = 24 - 8 * p;
        s_hist[t] = 0;
        __syncthreads();
        unsigned pref  = sh_prefix;
        unsigned kmask = (p == 0) ? 0u : (0xFFFFFFFFu << (32 - 8 * p));
        for (int i = t; i < A; i += TPB) {
            unsigned vv = s_cneg[i];
            if ((vv & kmask) == pref) atomicAdd(&s_hist[(vv >> shift) & 255], 1);
        }
        __syncthreads();
        if (t == 0) {
            int kk = sh_kk, cum = 0;
            for (int b2 = 255; b2 >= 0; --b2) {
                int h = s_hist[b2];
                if (cum + h >= kk) {
                    sh_G     += cum;
                    sh_kk     = kk - cum;
                    sh_prefix = pref | ((unsigned)b2 << shift);
                    break;
                }
                cum += h;
            }
        }
        __syncthreads();
    }
    unsigned T = sh_prefix;     // exact bit pattern of threshold value
    int      R = sh_kk;         // number of ties (value == T) to select

    // sum of con over elements strictly greater than T (all are negatives)
    float sg = 0.0f;
    for (int i = t; i < A; i += TPB)
        if (s_cneg[i] > T) sg += __uint_as_float(s_cneg[i]);
    float sumG = blockReduceSum(sg, s_red);

    // ordered tie scan: first R anchors (by index) with con_neg bits == T.
    // (Reproduces stable double-argsort, incl. positives doubled when T==0.)
    if (t == 0) sh_run = 0;
    __syncthreads();
    float st = 0.0f;
    for (int base = 0; base < A; base += TPB) {
        int  i    = base + t;
        int  run0 = sh_run;
        bool tie  = (i < A) && (s_cneg[i] == T);
        s_scan[t] = tie ? 1 : 0;
        __syncthreads();
        for (int o = 1; o < TPB; o <<= 1) {   // Hillis-Steele inclusive scan
            int vprev = (t >= o) ? s_scan[t - o] : 0;
            __syncthreads();
            s_scan[t] += vprev;
            __syncthreads();
        }
        int excl = s_scan[t] - (tie ? 1 : 0);
        if (tie && (run0 + excl) < R) st += s_con[i];
        int chunk_total = s_scan[TPB - 1];
        __syncthreads();
        if (t == 0) sh_run = run0 + chunk_total;
        __syncthreads();
    }
    float sumTie = blockReduceSum(st, s_red);

    if (t == 0) {
        float inv = 1.0f / (float)posn;
        ws_locc[n] = sh_loc * inv;
        ws_conc[n] = (sum_pos + sumG + sumTie) * inv;
    }
}

// ---------------------------------------------------------------------------
// Kernel C: task2 soft-label CE + final combine -> scalar
__global__ void final_kernel(const float* __restrict__ ws_locc,
                             const float* __restrict__ ws_conc,
                             const float* __restrict__ ptask2,
                             const float* __restrict__ gtask2,
                             float* __restrict__ out, int N) {
    __shared__ float r1[128], r2[128], r3[128];
    int t = threadIdx.x;
    float a = 0.0f, b = 0.0f, c = 0.0f;
    for (int i = t; i < N; i += 128) {
        a += ws_locc[i];
        b += ws_conc[i];
        float p0 = ptask2[2 * i], p1 = ptask2[2 * i + 1];
        float mm = fmaxf(p0, p1);
        float lse = mm + __logf(__expf(p0 - mm) + __expf(p1 - mm));
        float g0 = gtask2[2 * i], g1 = gtask2[2 * i + 1];
        c += -(g0 * (p0 - lse) + g1 * (p1 - lse));
    }
    r1[t] = a; r2[t] = b; r3[t] = c;
    __syncthreads();
    for (int o = 64; o > 0; o >>= 1) {
        if (t < o) { r1[t] += r1[t + o]; r2[t] += r2[t + o]; r3[t] += r3[t + o]; }
        __syncthreads();
    }
    if (t == 0) {
        float invN = 1.0f / (float)N;
        out[0] = 0.5f * ((r1[0] + r2[0]) * invN) + 0.5f * (r3[0] * invN);
    }
}

// ---------------------------------------------------------------------------
extern "C" void kernel_launch(void* const* d_in, const int* in_sizes, int n_in,
                              void* d_out, int out_size, void* d_ws, size_t ws_size,
                              hipStream_t stream) {
    const float* ploc   = (const float*)d_in[0];
    const float* plabel = (const float*)d_in[1];
    const float* gloc   = (const float*)d_in[2];
    const int*   glabel = (const int*)  d_in[3];
    const float* ptask2 = (const float*)d_in[4];
    const float* gtask2 = (const float*)d_in[5];
    const float* dboxes = (const float*)d_in[6];

    int A = in_sizes[6] / 4;                    // dboxes: [1,4,A]
    int N = in_sizes[4] / 2;                    // ptask2: [N,2]
    int C = (int)((long long)in_sizes[1] / ((long long)N * A));

    float* ws_con  = (float*)d_ws;              // N*A floats
    float* ws_locc = ws_con + (size_t)N * A;    // N floats
    float* ws_conc = ws_locc + N;               // N floats

    dim3 gA((unsigned)N, (unsigned)((A + TPB * 4 - 1) / (TPB * 4)));
    con_kernel<<<gA, TPB, 0, stream>>>(plabel, glabel, ws_con, N, C, A);

    size_t smem = (size_t)A * 8;                // con floats + con_neg bits (~70KB, <320KB)
    image_kernel<<<dim3((unsigned)N), TPB, smem, stream>>>(
        ploc, gloc, glabel, dboxes, ws_con, ws_locc, ws_conc, N, A);

    final_kernel<<<1, 128, 0, stream>>>(ws_locc, ws_conc, ptask2, gtask2,
                                        (float*)d_out, N);
}